// TensorMemory_25752623907456
// MI455X (gfx1250) — compile-verified
//
#include <hip/hip_runtime.h>
#include <stdint.h>

// ---------------------------------------------------------------------------
// CDNA5 async global<->LDS data-mover path (gfx1250). Guarded so the file
// still compiles if the toolchain lacks the builtins; fallback is plain b128.
// ---------------------------------------------------------------------------
#ifndef __has_builtin
#define __has_builtin(x) 0
#endif

#if __has_builtin(__builtin_amdgcn_global_load_async_to_lds_b128) && \
    __has_builtin(__builtin_amdgcn_global_store_async_from_lds_b128) && \
    __has_builtin(__builtin_amdgcn_s_wait_asynccnt)
#define USE_ASYNC_LDS 1
#else
#define USE_ASYNC_LDS 0
#endif

#if USE_ASYNC_LDS
// Builtin signature (from hipcc diagnostic): first param is
// 'int __vector(4) __device__ *' (AS1 global int4*), second is the LDS int4*.
typedef int v4i __attribute__((ext_vector_type(4)));
typedef __attribute__((address_space(1))) v4i* g_v4i_p;  // global
typedef __attribute__((address_space(3))) v4i* l_v4i_p;  // LDS

__device__ __forceinline__ void async_g2l_b128(const void* gsrc, void* lds) {
  // per-lane: copies 16 bytes from global address gsrc to LDS address lds, tracked by ASYNCcnt
  __builtin_amdgcn_global_load_async_to_lds_b128(
      (g_v4i_p)(uintptr_t)gsrc, (l_v4i_p)(uintptr_t)lds, 0, 0);
}
__device__ __forceinline__ void async_l2g_b128(void* gdst, const void* lds) {
  __builtin_amdgcn_global_store_async_from_lds_b128(
      (g_v4i_p)(uintptr_t)gdst, (l_v4i_p)(uintptr_t)lds, 0, 0);
}
__device__ __forceinline__ void wait_async0() {
  __builtin_amdgcn_s_wait_asynccnt(0);
}
#endif

// ---------------------------------------------------------------------------
// Kernels
// ---------------------------------------------------------------------------

__global__ void fill_i32_kernel(int* __restrict__ p, int v, int n) {
  int i = blockIdx.x * blockDim.x + threadIdx.x;
  if (i < n) p[i] = v;
}

// winner[node_idxs[i]] = max over i  -> "last occurrence wins" scatter
__global__ void winner_max_kernel(const int* __restrict__ idxs,
                                  int* __restrict__ winner, int B) {
  int i = blockIdx.x * blockDim.x + threadIdx.x;
  if (i < B) atomicMax(&winner[idxs[i]], i);
}

// One wave32 per gathered row (MEM_D = 128 floats = 32 float4, one per lane).
__global__ void gather_rows_kernel(const float* __restrict__ memory,
                                   const float* __restrict__ last_update,
                                   const int* __restrict__ idxs,
                                   float* __restrict__ out_mem,
                                   float* __restrict__ out_lu,
                                   int B, int d4 /* MEM_D/4 */) {
  int w = (blockIdx.x * blockDim.x + threadIdx.x) >> 5;  // wave id
  int lane = threadIdx.x & 31;
  if (w >= B) return;
  long idx = idxs[w];
  const float4* src = (const float4*)memory + idx * d4;
  float4* dst = (float4*)out_mem + (long)w * d4;
  for (int c = lane; c < d4; c += 32) dst[c] = src[c];
  if (lane == 0) out_lu[w] = last_update[idx];
}

// One wave32 per scattered row; only the last occurrence of an index writes.
__global__ void scatter_mem_kernel(const float* __restrict__ values,
                                   const int* __restrict__ idxs,
                                   const int* __restrict__ winner,
                                   float* __restrict__ out_newm,
                                   int B, int d4) {
  int w = (blockIdx.x * blockDim.x + threadIdx.x) >> 5;
  int lane = threadIdx.x & 31;
  if (w >= B) return;
  long idx = idxs[w];
  if (winner[idx] != w) return;  // not the winning update for this row
  const float4* src = (const float4*)values + (long)w * d4;
  float4* dst = (float4*)out_newm + idx * d4;
  for (int c = lane; c < d4; c += 32) dst[c] = src[c];
}

// messages row scatter: out[nodes[i]] = node_id_to_messages[nodes[i]]
// Value depends only on the row index, so duplicate writes are benign.
// Rows are 257 f32 (4B-aligned only) -> coalesced b32 accesses.
__global__ void scatter_msgs_kernel(const float* __restrict__ nim,
                                    const int* __restrict__ nodes,
                                    float* __restrict__ out_msgs,
                                    int B, int C) {
  int w = (blockIdx.x * blockDim.x + threadIdx.x) >> 5;
  int lane = threadIdx.x & 31;
  if (w >= B) return;
  long r = nodes[w];
  const float* src = nim + r * C;
  float* dst = out_msgs + r * C;
  for (int j = lane; j < C; j += 32) dst[j] = src[j];
}

// Bulk flat copy (float4 granularity). On gfx1250 with the async builtins we
// stage through LDS with GLOBAL_LOAD_ASYNC_TO_LDS_B128 /
// GLOBAL_STORE_ASYNC_FROM_LDS_B128 (ASYNCcnt + s_wait_asynccnt).
constexpr int CPY_BLOCK = 256;  // 8 wave32
constexpr int CPY_U = 4;        // float4 per thread per tile
constexpr int CPY_TILE = CPY_BLOCK * CPY_U;  // 1024 float4 = 16 KB LDS

__global__ void flat_copy_f4_kernel(const float4* __restrict__ src,
                                    float4* __restrict__ dst, long n4) {
#if USE_ASYNC_LDS
  __shared__ float4 buf[CPY_TILE];
  const long ntiles = n4 / CPY_TILE;
  for (long t = blockIdx.x; t < ntiles; t += gridDim.x) {
    const long base = t * (long)CPY_TILE;
#pragma unroll
    for (int u = 0; u < CPY_U; ++u) {
      const int s = u * CPY_BLOCK + threadIdx.x;  // unique slot per thread
      async_g2l_b128(&src[base + s], &buf[s]);
    }
    wait_async0();  // per-wave: our lanes' LDS slots are filled
#pragma unroll
    for (int u = 0; u < CPY_U; ++u) {
      const int s = u * CPY_BLOCK + threadIdx.x;
      async_l2g_b128(&dst[base + s], &buf[s]);
    }
    wait_async0();  // stores drained before LDS slots are reused
  }
  // tail (not a full tile): direct copy
  const long tail = ntiles * (long)CPY_TILE;
  for (long i = tail + (long)blockIdx.x * blockDim.x + threadIdx.x; i < n4;
       i += (long)gridDim.x * blockDim.x)
    dst[i] = src[i];
#else
  for (long i = (long)blockIdx.x * blockDim.x + threadIdx.x; i < n4;
       i += (long)gridDim.x * blockDim.x)
    dst[i] = src[i];
#endif
}

// ---------------------------------------------------------------------------
// Launch
// ---------------------------------------------------------------------------
extern "C" void kernel_launch(void* const* d_in, const int* in_sizes, int n_in,
                              void* d_out, int out_size, void* d_ws,
                              size_t ws_size, hipStream_t stream) {
  const float* memory      = (const float*)d_in[0];  // [N, MEM_D]
  const float* last_update = (const float*)d_in[1];  // [N]
  const float* messages    = (const float*)d_in[2];  // [N, MSG_D+1]
  const float* nim         = (const float*)d_in[3];  // [N, MSG_D+1]
  const float* values      = (const float*)d_in[4];  // [B, MEM_D]
  const int*   node_idxs   = (const int*)d_in[5];    // [B]
  const int*   nodes       = (const int*)d_in[6];    // [B]

  const long N    = in_sizes[1];                 // 500,000
  const long B    = in_sizes[5];                 // 200,000
  const int  MEMD = (int)(in_sizes[0] / N);      // 128
  const int  MSGC = (int)(in_sizes[2] / N);      // 257
  const int  D4   = MEMD / 4;                    // 32 float4 per row

  // Output layout: mem_out | lu_out | new_memory | new_messages
  float* out_mem  = (float*)d_out;
  float* out_lu   = out_mem + B * (long)MEMD;
  float* out_newm = out_lu + B;
  float* out_msgs = out_newm + N * (long)MEMD;

  // winner scratch (N ints). Prefer d_ws; otherwise borrow the head of the
  // new_messages output region (it is fully overwritten by the messages copy,
  // which runs strictly after the memory scatter on this stream).
  int* winner = (ws_size >= (size_t)N * sizeof(int)) ? (int*)d_ws
                                                     : (int*)out_msgs;

  const int TB = 256;
  const long rowThreads = B * 32;  // one wave32 per row

  // 1) winner = -1
  fill_i32_kernel<<<(int)((N + TB - 1) / TB), TB, 0, stream>>>(winner, -1,
                                                               (int)N);
  // 2) winner[idx] = max(i)  (last occurrence wins)
  winner_max_kernel<<<(int)((B + TB - 1) / TB), TB, 0, stream>>>(node_idxs,
                                                                 winner,
                                                                 (int)B);
  // 3) mem_out / lu_out gathers
  gather_rows_kernel<<<(int)((rowThreads + TB - 1) / TB), TB, 0, stream>>>(
      memory, last_update, node_idxs, out_mem, out_lu, (int)B, D4);
  // 4) new_memory = memory (bulk copy, async-LDS path)
  flat_copy_f4_kernel<<<2048, CPY_BLOCK, 0, stream>>>(
      (const float4*)memory, (float4*)out_newm, N * (long)MEMD / 4);
  // 5) new_memory[idx] = values[i] for winning i
  scatter_mem_kernel<<<(int)((rowThreads + TB - 1) / TB), TB, 0, stream>>>(
      values, node_idxs, winner, out_newm, (int)B, D4);
  // 6) new_messages = messages (bulk copy; overwrites borrowed winner space)
  flat_copy_f4_kernel<<<4096, CPY_BLOCK, 0, stream>>>(
      (const float4*)messages, (float4*)out_msgs, N * (long)MSGC / 4);
  // 7) new_messages[r] = node_id_to_messages[r] for r in nodes
  scatter_msgs_kernel<<<(int)((rowThreads + TB - 1) / TB), TB, 0, stream>>>(
      nim, nodes, out_msgs, (int)B, MSGC);
}